// ConstraintGenerativeModel_43490838840056
// MI455X (gfx1250) — compile-verified
//
#include <hip/hip_runtime.h>
#include <math.h>

// ---- problem dims (from reference) ----
#define BSZ  8192
#define TSZ  16
#define XUD  36
#define CTXD 16
#define KD   4
#define HD   32
#define BT   (BSZ*TSZ)          // 131072 samples
#define XIN  (XUD + CTXD)       // 52 input features

#define INV_C  (1.0f / (float)(BT * KD))        // 1/524288
#define INV_G  (1.0f / (float)(BT * KD * XUD))  // 1/18874368
#define INV_KL (1.0f / (float)(BSZ * CTXD))     // 1/131072

typedef float v2f __attribute__((ext_vector_type(2)));
typedef float v8f __attribute__((ext_vector_type(8)));

__device__ __forceinline__ v8f wmma4(v2f a, v2f b, v8f c) {
  // D(16x16,f32) = A(16x4,f32) x B(4x16,f32) + C
  return __builtin_amdgcn_wmma_f32_16x16x4_f32(
      /*neg_a=*/false, a, /*neg_b=*/false, b,
      /*c_mod=*/(short)0, c, /*reuse_a=*/false, /*reuse_b=*/false);
}

// ---------------- workspace init ----------------
__global__ void cgm_init_ws(float* ws) { ws[0] = 0.0f; }

// ---------------- context + KL ----------------
__global__ __launch_bounds__(256)
void cgm_ctx_kernel(const float* __restrict__ mu, const float* __restrict__ sg,
                    const float* __restrict__ no, float* __restrict__ ctx_out,
                    float* __restrict__ ws) {
  int idx = blockIdx.x * 256 + threadIdx.x;   // 0 .. B*CTX-1
  float m = mu[idx], s = sg[idx], n = no[idx];
  ctx_out[idx] = m + s * n;                   // reparameterized sample
  float kl = -logf(s) + 0.5f * (s * s + m * m) - 0.5f;
  float acc = kl * INV_KL;
  #pragma unroll
  for (int off = 16; off > 0; off >>= 1) acc += __shfl_down(acc, off, 32);
  if ((threadIdx.x & 31) == 0) atomicAdd(ws, acc);
}

// ---------------- main: MLP forward (WMMA) + analytic Jacobian + losses ----
__global__ __launch_bounds__(128)
void cgm_main_kernel(const float* __restrict__ traj, const float* __restrict__ tc,
                     const float* __restrict__ tg,   const float* __restrict__ ctx,
                     const float* __restrict__ W1,   const float* __restrict__ b1,
                     const float* __restrict__ W2,   const float* __restrict__ b2,
                     const float* __restrict__ W3,   const float* __restrict__ b3,
                     float* __restrict__ ws) {
  __shared__ float W1s[XIN][HD];   // W1 row-major: W1s[x][j]
  __shared__ float W2t[HD][HD];    // transposed:   W2t[h][j] = W2[j][h]
  __shared__ float W3p[HD][16];    // zero-padded:  W3p[h][0..3]=W3, 4..15 = 0
  __shared__ float b1s[HD], b2s[HD], b3s[16];
  __shared__ float zs1[4][32][33]; // per-wave Z1 (pre-ReLU), padded stride
  __shared__ float zs2[4][32][33]; // per-wave Z2 (pre-ReLU)

  const int tid = threadIdx.x;
  for (int i = tid; i < XIN * HD; i += 128) W1s[i >> 5][i & 31] = W1[i];
  for (int i = tid; i < HD * HD;  i += 128) W2t[i & 31][i >> 5] = W2[i];
  for (int i = tid; i < HD * 16;  i += 128) {
    int h = i >> 4, n = i & 15;
    W3p[h][n] = (n < KD) ? W3[h * KD + n] : 0.0f;
  }
  if (tid < HD) { b1s[tid] = b1[tid]; b2s[tid] = b2[tid]; }
  if (tid < 16) { b3s[tid] = (tid < KD) ? b3[tid] : 0.0f; }
  __syncthreads();

  const int lane = tid & 31;
  const int w    = tid >> 5;        // wave in block (0..3)
  const int hi   = lane >> 4;       // 0/1: K-pair / M-row-half selector
  const int lo   = lane & 15;
  const int s0   = (blockIdx.x * 4 + w) * 32;  // first sample of this wave

  float acc = 0.0f;

  // ---- forward pass, two 16-row tiles per wave ----
  for (int rt = 0; rt < 2; ++rt) {
    const int srow = s0 + rt * 16;
    const int bb   = srow >> 4;     // batch index (uniform over tile, T==16)

    // layer 1: Z1(16x32) = X(16x52) @ W1(52x32) + b1   (13 K-chunks, 2 N-tiles)
    v8f c0 = {}; v8f c1 = {};
    for (int kc = 0; kc < 13; ++kc) {
      const int kb = kc * 4;             // uniform chunk base
      const int ka = kb + 2 * hi;        // lane's K pair
      v2f a;
      if (kb < XUD) {                    // whole chunk from trajectories
        const float* xr = traj + (size_t)(srow + lo) * XUD + ka;
        a.x = xr[0]; a.y = xr[1];
      } else {                           // whole chunk from context (row-bcast)
        const float* cr = ctx + (size_t)bb * CTXD + (ka - XUD);
        a.x = cr[0]; a.y = cr[1];
      }
      v2f p; p.x = W1s[ka][lo];      p.y = W1s[ka + 1][lo];
      v2f q; q.x = W1s[ka][16 + lo]; q.y = W1s[ka + 1][16 + lo];
      c0 = wmma4(a, p, c0);
      c1 = wmma4(a, q, c1);
    }
    #pragma unroll
    for (int r = 0; r < 8; ++r) {
      int row = rt * 16 + r + 8 * hi;
      zs1[w][row][lo]      = c0[r] + b1s[lo];
      zs1[w][row][16 + lo] = c1[r] + b1s[16 + lo];
    }

    // layer 2: Z2 = ReLU(Z1) @ W2 + b2   (8 K-chunks, 2 N-tiles)
    v8f d0 = {}; v8f d1 = {};
    for (int kc = 0; kc < 8; ++kc) {
      const int ka = kc * 4 + 2 * hi;
      float z0 = zs1[w][rt * 16 + lo][ka];
      float z1v = zs1[w][rt * 16 + lo][ka + 1];
      v2f a; a.x = (z0 > 0.0f) ? z0 : 0.0f; a.y = (z1v > 0.0f) ? z1v : 0.0f;
      v2f p; p.x = W2t[lo][ka];      p.y = W2t[lo][ka + 1];
      v2f q; q.x = W2t[16 + lo][ka]; q.y = W2t[16 + lo][ka + 1];
      d0 = wmma4(a, p, d0);
      d1 = wmma4(a, q, d1);
    }
    #pragma unroll
    for (int r = 0; r < 8; ++r) {
      int row = rt * 16 + r + 8 * hi;
      zs2[w][row][lo]      = d0[r] + b2s[lo];
      zs2[w][row][16 + lo] = d1[r] + b2s[16 + lo];
    }

    // layer 3: pred(16x4) = ReLU(Z2) @ W3 + b3 (one N-tile, cols 4..15 zero)
    v8f e = {};
    for (int kc = 0; kc < 8; ++kc) {
      const int ka = kc * 4 + 2 * hi;
      float z0 = zs2[w][rt * 16 + lo][ka];
      float z1v = zs2[w][rt * 16 + lo][ka + 1];
      v2f a; a.x = (z0 > 0.0f) ? z0 : 0.0f; a.y = (z1v > 0.0f) ? z1v : 0.0f;
      v2f p; p.x = W3p[ka][lo]; p.y = W3p[ka + 1][lo];
      e = wmma4(a, p, e);
    }
    const int ci = (lo < KD) ? lo : 0;   // clamp to keep loads in-bounds
    #pragma unroll
    for (int r = 0; r < 8; ++r) {
      int m = rt * 16 + r + 8 * hi;
      float p  = e[r] + b3s[lo];
      float t  = tc[(size_t)(s0 + m) * KD + ci];
      float df = p - t;
      acc += (lo < KD) ? df * df * INV_C : 0.0f;
    }
  }

  // ---- analytic Jacobian, one lane per sample ----
  const int s = s0 + lane;
  const float* zr1 = &zs1[w][lane][0];
  const float* zr2 = &zs2[w][lane][0];
  const float* tgr = tg + (size_t)s * (KD * XUD);  // 144 contiguous floats/lane
  __builtin_prefetch(tgr, 0, 1);

  #pragma unroll 1
  for (int k = 0; k < KD; ++k) {
    float c[HD];
    #pragma unroll
    for (int j = 0; j < HD; ++j) c[j] = 0.0f;
    // c[j] = sum_h W2[j,h] * m2[h] * W3[h,k]
    #pragma unroll 1
    for (int h = 0; h < HD; ++h) {
      float z2 = zr2[h];
      float g  = (z2 > 0.0f) ? W3p[h][k] : 0.0f;
      #pragma unroll
      for (int j = 0; j < HD; ++j) c[j] = fmaf(W2t[h][j], g, c[j]);
    }
    #pragma unroll
    for (int j = 0; j < HD; ++j) c[j] = (zr1[j] > 0.0f) ? c[j] : 0.0f;

    // J[k,x] = sum_j W1[x,j] * c[j] ; targets read as b128 (16B-aligned:
    // row offset = s*576 + k*144 bytes). 9 blocks of 4 columns.
    const float4* tg4 = (const float4*)(tgr + k * XUD);
    #pragma unroll 1
    for (int q = 0; q < (XUD / 4); ++q) {
      float4 tv = tg4[q];
      float tvv[4] = {tv.x, tv.y, tv.z, tv.w};
      #pragma unroll
      for (int xi = 0; xi < 4; ++xi) {
        const float* w1r = &W1s[q * 4 + xi][0];
        float jx = 0.0f;
        #pragma unroll
        for (int j = 0; j < HD; ++j) jx = fmaf(w1r[j], c[j], jx);
        float df = jx - tvv[xi];
        acc += df * df * INV_G;
      }
    }
  }

  // ---- wave reduction + one atomic per wave ----
  #pragma unroll
  for (int off = 16; off > 0; off >>= 1) acc += __shfl_down(acc, off, 32);
  if (lane == 0) atomicAdd(ws, acc);
}

// ---------------- finalize ----------------
__global__ void cgm_finalize(const float* ws, float* out) { out[0] = ws[0]; }

extern "C" void kernel_launch(void* const* d_in, const int* in_sizes, int n_in,
                              void* d_out, int out_size, void* d_ws, size_t ws_size,
                              hipStream_t stream) {
  const float* traj = (const float*)d_in[0];
  const float* tc   = (const float*)d_in[1];
  const float* tg   = (const float*)d_in[2];
  const float* mu   = (const float*)d_in[3];
  const float* sg   = (const float*)d_in[4];
  const float* no   = (const float*)d_in[5];
  const float* W1   = (const float*)d_in[6];
  const float* b1   = (const float*)d_in[7];
  const float* W2   = (const float*)d_in[8];
  const float* b2   = (const float*)d_in[9];
  const float* W3   = (const float*)d_in[10];
  const float* b3   = (const float*)d_in[11];
  float* out = (float*)d_out;   // [0]=loss, [1..]=context (B,CTX)
  float* ws  = (float*)d_ws;    // [0]=loss accumulator

  cgm_init_ws<<<1, 1, 0, stream>>>(ws);
  cgm_ctx_kernel<<<(BSZ * CTXD) / 256, 256, 0, stream>>>(mu, sg, no, out + 1, ws);
  cgm_main_kernel<<<BT / 128, 128, 0, stream>>>(traj, tc, tg, out + 1,
                                                W1, b1, W2, b2, W3, b3, ws);
  cgm_finalize<<<1, 1, 0, stream>>>(ws, out);
}